// LeakyRNN_26749056319796
// MI455X (gfx1250) — compile-verified
//
#include <hip/hip_runtime.h>
#include <hip/hip_bf16.h>
#include <math.h>

// ---------------------------------------------------------------------------
// Leaky softplus RNN on MI455X (gfx1250, wave32).
//   E = 512 expanded rows, H = 512 hidden, 100 warmup + 400 collect steps.
//   Per step: rec = softplus(h) @ W_rec^T   (512x512x512 fp32 GEMM via
//   v_wmma_f32_16x16x4_f32), h = 0.9*h + 0.1*(rec + u), fused output
//   projection + atomics for collect steps. Ping-pong h buffers per step.
// ---------------------------------------------------------------------------

#define HIDDEN 512
#define INPUT 64
#define EXPANDED 512          // batch(64) * trials(8)
#define WARMUP 100
#define RESPONSE 400
#define NTRIALS 8
#define ALPHA 0.1f
#define ONE_MINUS_ALPHA 0.9f

typedef float v2f __attribute__((ext_vector_type(2)));
typedef float v8f __attribute__((ext_vector_type(8)));

__device__ __forceinline__ float softplus_f(float x) {
    // jax.nn.softplus = log1p(exp(x)); safe for large |x|
    return (x > 20.0f) ? x : log1pf(expf(x));
}

// ---------------- utility: fill buffer ----------------
__global__ void fill_kernel(float* __restrict__ p, int n, float v) {
    int t = blockIdx.x * blockDim.x + threadIdx.x;
    if (t < n) p[t] = v;
}

// ---------------- input contribution: u[e][j] = b_in[j] + x[e] . W_in[j] ----
__global__ void input_contrib_kernel(const float* __restrict__ inputs,
                                     const float* __restrict__ W_in,
                                     const float* __restrict__ b_in,
                                     float* __restrict__ u) {
    int t = blockIdx.x * blockDim.x + threadIdx.x;   // 0 .. E*H-1
    if (t >= EXPANDED * HIDDEN) return;
    int e = t >> 9;          // / HIDDEN
    int j = t & (HIDDEN - 1);
    const float* xrow = inputs + (e >> 3) * INPUT;   // row e/num_trials
    const float* wrow = W_in + j * INPUT;
    float acc = b_in[j];
#pragma unroll 8
    for (int k = 0; k < INPUT; ++k) acc += xrow[k] * wrow[k];
    u[t] = acc;
}

// ---------------- per-timestep kernel ----------------
// Grid: (8, 8) blocks of 256 threads (8 wave32s). Block computes 64x64 tile
// of rec = softplus(hin) @ W_rec^T, then the leaky update into hout.
// Wave w: wr = w&3 -> 16-row strip, wc = w>>2 -> 32-col strip (two 16x16 D tiles).
__global__ void __launch_bounds__(256)
rnn_step_kernel(const float* __restrict__ hin,
                float* __restrict__ hout,
                const float* __restrict__ u,
                const float* __restrict__ W_rec,
                const float* __restrict__ w_out,
                float* __restrict__ out_accum,
                int sidx) {                 // sidx = step - WARMUP (<0: no collect)
    __shared__ float A_s[64][132];          // softplus(h) tile, +4 pad for banks
    __shared__ float B_s[64][132];          // W_rec rows (B^T tile)

    const int rowBase = blockIdx.y * 64;
    const int colBase = blockIdx.x * 64;
    const int wave = threadIdx.x >> 5;
    const int lane = threadIdx.x & 31;
    const int wr = wave & 3;                // row strip within block
    const int wc = wave >> 2;               // col strip within block

    v8f acc0 = {};
    v8f acc1 = {};

    const int mn   = lane & 15;             // A row / B col within 16
    const int ksel = (lane >> 4) << 1;      // K offset 0 or 2 (ISA A/B layout)

    for (int kc = 0; kc < HIDDEN; kc += 128) {
        // cooperative stage: A = softplus(h rows), B = W_rec rows
        for (int i = threadIdx.x; i < 64 * 128; i += 256) {
            int r = i >> 7, c = i & 127;
            A_s[r][c] = softplus_f(hin[(rowBase + r) * HIDDEN + kc + c]);
            B_s[r][c] = W_rec[(colBase + r) * HIDDEN + kc + c];
        }
        __syncthreads();

#pragma unroll 4
        for (int kk = 0; kk < 128; kk += 4) {
            v2f a  = *(const v2f*)&A_s[wr * 16 + mn][kk + ksel];
            v2f b0 = *(const v2f*)&B_s[wc * 32 + mn][kk + ksel];
            v2f b1 = *(const v2f*)&B_s[wc * 32 + 16 + mn][kk + ksel];
            acc0 = __builtin_amdgcn_wmma_f32_16x16x4_f32(
                false, a, false, b0, (short)0, acc0, false, false);
            acc1 = __builtin_amdgcn_wmma_f32_16x16x4_f32(
                false, a, false, b1, (short)0, acc1, false, false);
        }
        __syncthreads();
    }

    // Epilogue: leaky update + fused output projection.
    // D layout: VGPR i -> rows (i, i+8); lanes 0-15 / 16-31 are the two halves,
    // N = lane & 15. All 16 lanes of a half share the same row.
    const int halfsel = (lane >> 4) << 3;   // row offset 0 or 8
    for (int t = 0; t < 2; ++t) {
        const v8f acc = t ? acc1 : acc0;
        const int col = colBase + wc * 32 + t * 16 + (lane & 15);
        const float wv = w_out[col];
#pragma unroll
        for (int i = 0; i < 8; ++i) {
            const int row = rowBase + wr * 16 + i + halfsel;
            const int idx = row * HIDDEN + col;
            const float hnew = ONE_MINUS_ALPHA * hin[idx]
                             + ALPHA * (acc[i] + u[idx]);
            hout[idx] = hnew;
            if (sidx >= 0) {
                float p = hnew * wv;
                p += __shfl_xor(p, 1, 16);
                p += __shfl_xor(p, 2, 16);
                p += __shfl_xor(p, 4, 16);
                p += __shfl_xor(p, 8, 16);
                if ((lane & 15) == 0)
                    atomicAdd(&out_accum[sidx * EXPANDED + row], p);
            }
        }
    }
}

// ---------------- finalize: sigmoid + transpose to [e][s] ----------------
__global__ void finalize_kernel(const float* __restrict__ out_accum,
                                const float* __restrict__ b_out,
                                float* __restrict__ out) {
    int t = blockIdx.x * blockDim.x + threadIdx.x;   // 0 .. E*RESPONSE-1
    if (t >= EXPANDED * RESPONSE) return;
    int e = t / RESPONSE;
    int s = t - e * RESPONSE;
    float x = out_accum[s * EXPANDED + e] + b_out[0];
    out[t] = 1.0f / (1.0f + expf(-x));
}

extern "C" void kernel_launch(void* const* d_in, const int* in_sizes, int n_in,
                              void* d_out, int out_size, void* d_ws, size_t ws_size,
                              hipStream_t stream) {
    const float* inputs = (const float*)d_in[0];   // [64,64]
    const float* W_in   = (const float*)d_in[1];   // [512,64]
    const float* b_in   = (const float*)d_in[2];   // [512]
    const float* W_rec  = (const float*)d_in[3];   // [512,512]
    const float* w_out  = (const float*)d_in[4];   // [1,512]
    const float* b_out  = (const float*)d_in[5];   // [1]
    float* out = (float*)d_out;                    // [64*8*400] = e*400+s

    float* ws = (float*)d_ws;
    float* h0        = ws;                                // 512*512
    float* h1        = ws + EXPANDED * HIDDEN;            // 512*512
    float* u         = ws + 2 * EXPANDED * HIDDEN;        // 512*512
    float* out_accum = ws + 3 * EXPANDED * HIDDEN;        // 400*512

    const int EH = EXPANDED * HIDDEN;          // 262144
    const int OA = RESPONSE * EXPANDED;        // 204800

    // h starts at zero; out_accum accumulated via atomics -> zero it
    fill_kernel<<<(EH + 255) / 256, 256, 0, stream>>>(h0, EH, 0.0f);
    fill_kernel<<<(OA + 255) / 256, 256, 0, stream>>>(out_accum, OA, 0.0f);

    input_contrib_kernel<<<(EH + 255) / 256, 256, 0, stream>>>(inputs, W_in, b_in, u);

    dim3 grid(HIDDEN / 64, EXPANDED / 64);     // 8 x 8 tiles
    const int TOTAL_STEPS = WARMUP + RESPONSE; // 500
    for (int s = 0; s < TOTAL_STEPS; ++s) {
        const float* hin = (s & 1) ? h1 : h0;
        float*       hot = (s & 1) ? h0 : h1;
        rnn_step_kernel<<<grid, 256, 0, stream>>>(hin, hot, u, W_rec, w_out,
                                                  out_accum, s - WARMUP);
    }

    finalize_kernel<<<(OA + 255) / 256, 256, 0, stream>>>(out_accum, b_out, out);
}